// LinODE_37349035606791
// MI455X (gfx1250) — compile-verified
//
#include <hip/hip_runtime.h>

// LinODE forward on gfx1250:
//   x_{t+1} = expm(dt_{t,b} * A) x_t,  A fixed 128x128 skew-symmetric.
// Computed as Taylor series of the *action*: per term a 128x128 @ 128x16 GEMM
// done with V_WMMA_F32_16X16X4_F32 (f32 in / f32 acc), batch = N dimension.
// Two phases: sequential checkpoint scan (1 block), then 16 parallel groups.
//
// Round 3 changes (from asm inspection):
//  - fully unroll the Taylor-term loop and use a constexpr 1/k table so the
//    per-term scale is a single v_mul with a literal (the rolled loop was
//    emitting a full precise-division sequence per term)

typedef __attribute__((ext_vector_type(2))) float v2f;
typedef __attribute__((ext_vector_type(8))) float v8f;

namespace {

constexpr int kB   = 16;    // batch (N dimension of WMMA)
constexpr int kLen = 128;   // time steps
constexpr int kD   = 128;   // state dim
constexpr int kNK  = 32;    // K-tiles of 4 along D (128/4)
constexpr int kXN  = kD * kB;  // floats per state matrix X[d][b]

// Exactly-rounded IEEE reciprocals, folded to literals in the unrolled loop.
constexpr float kInv[21] = {
    0.0f,        1.0f,        1.0f / 2,  1.0f / 3,  1.0f / 4,  1.0f / 5,
    1.0f / 6,    1.0f / 7,    1.0f / 8,  1.0f / 9,  1.0f / 10, 1.0f / 11,
    1.0f / 12,   1.0f / 13,   1.0f / 14, 1.0f / 15, 1.0f / 16, 1.0f / 17,
    1.0f / 18,   1.0f / 19,   1.0f / 20};

// Load this wave's A fragments: wave m owns rows 16m..16m+15.
// ISA 32-bit A 16x4 layout: lanes 0-15 -> {K=0,K=1}, lanes 16-31 -> {K=2,K=3},
// M = lane%16 in both halves.
__device__ __forceinline__ void load_a_frags(const float* __restrict__ A,
                                             int m, int lrow, int half,
                                             v2f (&af)[kNK]) {
  const float* base = A + (m * 16 + lrow) * kD + 2 * half;
#pragma unroll
  for (int kt = 0; kt < kNK; ++kt) {
    af[kt] = *(const v2f*)(base + 4 * kt);  // cols 4kt+2*half, +1 (8B aligned)
  }
}

// One exp step: X <- expm(tau*A) X via KTERMS Taylor terms.
// X lives in LDS as X[d*16 + b]; Xb/Yb ping-pong, Xb holds new state on exit.
template <int KTERMS>
__device__ __forceinline__ void taylor_step(const v2f (&af)[kNK],
                                            float*& Xb, float*& Yb,
                                            float tau, int m, int lrow,
                                            int half) {
  float* rd = Xb;
  float* wr = Yb;
  const int crow0 = m * 16 + half * 8;  // first C/D row this lane holds

  // S = P_0 = X (this wave's 16x16 tile, C-layout: VGPR r -> row crow0+r)
  v8f S;
#pragma unroll
  for (int r = 0; r < 8; ++r) S[r] = rd[(crow0 + r) * kB + lrow];

#pragma unroll
  for (int k = 1; k <= KTERMS; ++k) {
    // Batch-load ALL B fragments for this term: issues as one LDS clause,
    // letting the scheduler pipeline loads under the WMMAs.
    // B 4x16 f32 layout (mirrors A/C): vgpr0 = {K0 | K2}, vgpr1 = {K1 | K3},
    // N = lane%16. Bank-conflict-free: 32 lanes -> 32 distinct banks.
    v2f bf[kNK];
#pragma unroll
    for (int kt = 0; kt < kNK; ++kt) {
      const int base = (4 * kt + 2 * half) * kB + lrow;
      bf[kt].x = rd[base];
      bf[kt].y = rd[base + kB];
    }

    // 4 independent accumulators: dependency chain 32 -> 8 WMMAs.
    v8f a0 = {0.f, 0.f, 0.f, 0.f, 0.f, 0.f, 0.f, 0.f};
    v8f a1 = a0, a2 = a0, a3 = a0;
#pragma unroll
    for (int kt = 0; kt < kNK; kt += 4) {
      a0 = __builtin_amdgcn_wmma_f32_16x16x4_f32(false, af[kt + 0], false,
                                                 bf[kt + 0], (short)0, a0,
                                                 false, false);
      a1 = __builtin_amdgcn_wmma_f32_16x16x4_f32(false, af[kt + 1], false,
                                                 bf[kt + 1], (short)0, a1,
                                                 false, false);
      a2 = __builtin_amdgcn_wmma_f32_16x16x4_f32(false, af[kt + 2], false,
                                                 bf[kt + 2], (short)0, a2,
                                                 false, false);
      a3 = __builtin_amdgcn_wmma_f32_16x16x4_f32(false, af[kt + 3], false,
                                                 bf[kt + 3], (short)0, a3,
                                                 false, false);
    }

    // P_k = (tau/k) * A * P_{k-1}; k is a compile-time constant here, so the
    // scale is one v_mul by a literal (bit-identical to the precise division).
    const float f = tau * kInv[k];
    v8f acc;
#pragma unroll
    for (int r = 0; r < 8; ++r) {
      acc[r] = ((a0[r] + a1[r]) + (a2[r] + a3[r])) * f;
      S[r] += acc[r];
    }
    if (k < KTERMS) {
#pragma unroll
      for (int r = 0; r < 8; ++r) wr[(crow0 + r) * kB + lrow] = acc[r];
      __syncthreads();
      float* t = rd; rd = wr; wr = t;
    }
  }
  // Publish new state (wr was last read two terms ago; all reads fenced).
#pragma unroll
  for (int r = 0; r < 8; ++r) wr[(crow0 + r) * kB + lrow] = S[r];
  __syncthreads();
  Xb = wr;
  Yb = rd;
}

}  // namespace

// Phase 1: one block computes checkpoints C_g = expm((T[8g]-T[8g-8])A) C_{g-1}
// for g=1..15 (tau up to ~1.6 -> 20 Taylor terms for fp32 accuracy).
__global__ __launch_bounds__(256) void linode_phase1(
    const float* __restrict__ T, const float* __restrict__ x0,
    const float* __restrict__ A, float* __restrict__ Cws) {
  __shared__ float P0[kXN];
  __shared__ float P1[kXN];
  const int tid = threadIdx.x;
  const int m = tid >> 5, lane = tid & 31, half = lane >> 4, lrow = lane & 15;

  v2f af[kNK];
  load_a_frags(A, m, lrow, half, af);

  // Stage x0 (B,D row-major) -> X[d*16+b]
  for (int i = tid; i < kXN; i += 256) P0[i] = x0[(i & 15) * kD + (i >> 4)];
  __syncthreads();

  float* Xb = P0;
  float* Yb = P1;
  for (int g = 1; g <= 15; ++g) {
    const float tau = T[lrow * kLen + 8 * g] - T[lrow * kLen + 8 * (g - 1)];
    taylor_step<20>(af, Xb, Yb, tau, m, lrow, half);
    float* dst = Cws + (g - 1) * kXN;  // checkpoint, same [d*16+b] layout
    for (int i = tid; i < kXN; i += 256) dst[i] = Xb[i];
  }
}

// Phase 2: 16 blocks in parallel; block g starts from checkpoint C_g (x0 for
// g=0) and produces outputs t = 8g+1 .. 8g+8 (group 15: 7 steps).
__global__ __launch_bounds__(256) void linode_phase2(
    const float* __restrict__ T, const float* __restrict__ x0,
    const float* __restrict__ A, const float* __restrict__ Cws,
    float* __restrict__ out) {
  __shared__ float P0[kXN];
  __shared__ float P1[kXN];
  const int g = blockIdx.x;
  const int tid = threadIdx.x;
  const int m = tid >> 5, lane = tid & 31, half = lane >> 4, lrow = lane & 15;

  v2f af[kNK];
  load_a_frags(A, m, lrow, half, af);

  if (g == 0) {
    // Output row t=0 is x0 itself (coalesced: consecutive d per batch row).
    for (int i = tid; i < kXN; i += 256)
      out[(i >> 7) * (kLen * kD) + (i & 127)] = x0[i];
    for (int i = tid; i < kXN; i += 256)
      P0[i] = x0[(i & 15) * kD + (i >> 4)];
  } else {
    const float* src = Cws + (g - 1) * kXN;
    for (int i = tid; i < kXN; i += 256) P0[i] = src[i];
  }
  __syncthreads();

  float* Xb = P0;
  float* Yb = P1;
  const int nsteps = (g == 15) ? 7 : 8;
  for (int j = 1; j <= nsteps; ++j) {
    const int t = 8 * g + j;
    const float dt = T[lrow * kLen + t] - T[lrow * kLen + (t - 1)];
    taylor_step<9>(af, Xb, Yb, dt, m, lrow, half);
    // Coalesced store through the LDS state buffer: out[b][t][d] = X'[d][b]
    for (int i = tid; i < kXN; i += 256) {
      const int b = i >> 7, d = i & 127;
      out[b * (kLen * kD) + t * kD + d] = Xb[d * kB + b];
    }
  }
}

extern "C" void kernel_launch(void* const* d_in, const int* in_sizes, int n_in,
                              void* d_out, int out_size, void* d_ws,
                              size_t ws_size, hipStream_t stream) {
  const float* T  = (const float*)d_in[0];   // (B, LEN)
  const float* x0 = (const float*)d_in[1];   // (B, D)
  const float* A  = (const float*)d_in[2];   // (D, D) row-major
  float* out = (float*)d_out;                // (B, LEN, D)
  float* Cws = (float*)d_ws;                 // 15 checkpoints * 2048 f32 = 120KB

  linode_phase1<<<1, 256, 0, stream>>>(T, x0, A, Cws);
  linode_phase2<<<16, 256, 0, stream>>>(T, x0, A, Cws, out);
}